// LabelAwareGCN_19353122636455
// MI455X (gfx1250) — compile-verified
//
#include <hip/hip_runtime.h>
#include <hip/hip_bf16.h>

typedef float v2f __attribute__((ext_vector_type(2)));
typedef float v8f __attribute__((ext_vector_type(8)));

#define B_   8
#define L_   128
#define D_   768
#define DEP_ 64
#define POS_ 64
#define OUT_ 768
#define NEGV (-1e30f)

#define R_ELEMS  ((size_t)B_ * L_ * L_ * OUT_)   // 100663296
#define OS_ELEMS ((size_t)B_ * L_ * OUT_)        // 786432
#define P_ELEMS  ((size_t)B_ * L_ * L_)          // 131072

// Async global->LDS staging helpers (CDNA5 GLOBAL_LOAD_ASYNC_TO_LDS_*).
// LDS operand: low 32 bits of a generic LDS pointer are the LDS byte offset
__device__ __forceinline__ void async_copy_b128(void* lds_dst, const void* gsrc) {
    unsigned laddr = (unsigned)(size_t)lds_dst;
    unsigned long long g = (unsigned long long)(size_t)gsrc;
    asm volatile("global_load_async_to_lds_b128 %0, %1, off"
                 :: "v"(laddr), "v"(g) : "memory");
}
__device__ __forceinline__ void async_copy_b32(void* lds_dst, const void* gsrc) {
    unsigned laddr = (unsigned)(size_t)lds_dst;
    unsigned long long g = (unsigned long long)(size_t)gsrc;
    asm volatile("global_load_async_to_lds_b32 %0, %1, off"
                 :: "v"(laddr), "v"(g) : "memory");
}
__device__ __forceinline__ void wait_async() {
    asm volatile("s_wait_asynccnt 0" ::: "memory");
}

// ---------------------------------------------------------------------------
// Kernel 0: column sums of the attention weight blocks (for the scores trick)
// ws layout: [0..767]=wtsum, [768..831]=wpsum, [832..895]=wdsum, [896]=bsum
// ---------------------------------------------------------------------------
__global__ void colsum_kernel(const float* __restrict__ Wt,
                              const float* __restrict__ Wp_attn,
                              const float* __restrict__ Wd_attn,
                              const float* __restrict__ b_attn,
                              float* __restrict__ ws) {
    int t = threadIdx.x;  // 256 threads
    for (int d = t; d < D_; d += 256) {
        float s = 0.f;
        const float* row = Wt + (size_t)d * OUT_;
        for (int o = 0; o < OUT_; ++o) s += row[o];
        ws[d] = s;
    }
    if (t < DEP_) {
        float sp = 0.f, sd = 0.f;
        const float* rp = Wp_attn + (size_t)t * OUT_;
        const float* rd = Wd_attn + (size_t)t * OUT_;
        for (int o = 0; o < OUT_; ++o) { sp += rp[o]; sd += rd[o]; }
        ws[768 + t] = sp;
        ws[832 + t] = sd;
    }
    if (t == 0) {
        float s = 0.f;
        for (int o = 0; o < OUT_; ++o) s += b_attn[o];
        ws[896] = s;
    }
}

// ---------------------------------------------------------------------------
// Kernel 1: scores (via column sums) + softmax over j -> p  [one block per (b,i)]
// ---------------------------------------------------------------------------
__global__ void __launch_bounds__(128)
softmax_kernel(const float* __restrict__ text, const int* __restrict__ adj,
               const float* __restrict__ dep,  const float* __restrict__ pos,
               const float* __restrict__ ws,   float* __restrict__ p_out) {
    int bi = blockIdx.x;     // b*L + i
    int j  = threadIdx.x;    // 128 threads: one per neighbor j
    __shared__ float red[128];

    const float* wtsum = ws;
    const float* wpsum = ws + 768;
    const float* wdsum = ws + 832;
    float bsum = ws[896];

    float part = 0.f;
    for (int d = j; d < D_; d += 128) part += text[(size_t)bi * D_ + d] * wtsum[d];
    if (j < POS_) part += pos[(size_t)bi * POS_ + j] * wpsum[j];
    red[j] = part;
    __syncthreads();
    for (int s = 64; s > 0; s >>= 1) {
        if (j < s) red[j] += red[j + s];
        __syncthreads();
    }
    float base = red[0] + bsum;
    __syncthreads();

    const float* drow = dep + ((size_t)bi * L_ + j) * DEP_;
    float dot = 0.f;
    #pragma unroll
    for (int d = 0; d < DEP_; d += 4) {
        float4 v = *(const float4*)(drow + d);
        dot += v.x * wdsum[d] + v.y * wdsum[d + 1] + v.z * wdsum[d + 2] + v.w * wdsum[d + 3];
    }
    float score = base + dot + (adj[(size_t)bi * L_ + j] == 0 ? NEGV : 0.f);

    red[j] = score;
    __syncthreads();
    for (int s = 64; s > 0; s >>= 1) {
        if (j < s) red[j] = fmaxf(red[j], red[j + s]);
        __syncthreads();
    }
    float m = red[0];
    __syncthreads();
    float e = __expf(score - m);
    red[j] = e;
    __syncthreads();
    for (int s = 64; s > 0; s >>= 1) {
        if (j < s) red[j] += red[j + s];
        __syncthreads();
    }
    p_out[(size_t)bi * L_ + j] = e / red[0];
}

// ---------------------------------------------------------------------------
// Kernel 2 (WMMA): rowvec[m,o] = [text|pos][m,:] @ [Wt;Wp_attn] + b_attn
// GEMM M=1024, K=832, N=768 on the matrix pipe. B tile stored K-pair-swizzled
// in LDS so each lane's WMMA B fragment is one aligned ds_load_b64.
// ---------------------------------------------------------------------------
__global__ void __launch_bounds__(128)
rowvec_wmma_kernel(const float* __restrict__ text, const float* __restrict__ pos,
                   const float* __restrict__ Wt,   const float* __restrict__ Wp_attn,
                   const float* __restrict__ b_attn, float* __restrict__ rowvec) {
    int m0 = (blockIdx.x / 6) * 16;     // M tile (rows of text/pos, b*L flat)
    int n0 = (blockIdx.x % 6) * 128;    // N chunk
    int t    = threadIdx.x;             // 128
    int wave = t >> 5;
    int lane = t & 31;
    int mrow = lane & 15;
    int hi   = lane >> 4;

    __shared__ float sA[16 * 832];      // [row][k], k<768 text, k>=768 pos
    __shared__ float sB[64 * 128];      // swizzled: [(kk*2+h)*128 + n][pair s]

    // stage A: text rows m0..m0+15 (contiguous 12288 floats) + pos rows
    {
        const float4* tsrc = (const float4*)(text + (size_t)m0 * D_);
        for (int k = t; k < 3072; k += 128) {
            int row = k / 192, c4 = k % 192;
            *(float4*)&sA[row * 832 + c4 * 4] = tsrc[k];
        }
        const float4* psrc = (const float4*)(pos + (size_t)m0 * POS_);
        for (int k = t; k < 256; k += 128) {
            int row = k / 16, c4 = k % 16;
            *(float4*)&sA[row * 832 + 768 + c4 * 4] = psrc[k];
        }
    }
    __syncthreads();

    v8f acc0 = {}, acc1 = {};
    int nbase0 = wave * 16;
    int nbase1 = wave * 16 + 64;

    for (int c = 0; c < 13; ++c) {      // 13 K-chunks of 64 (chunk 12 = pos block)
        int kbase = c * 64;
        // stage B chunk swizzled: element B[d][n] (d local 0..63) at
        // sB[(kk*2+h)*256 + n*2 + s] where d = 4*kk + h + 2*s
        for (int idx = t; idx < 2048; idx += 128) {
            int d = idx >> 5, c4 = idx & 31;
            int k = kbase + d;
            const float* src = (k < D_) ? (Wt + (size_t)k * OUT_ + n0)
                                        : (Wp_attn + (size_t)(k - D_) * OUT_ + n0);
            float4 v = *(const float4*)(src + c4 * 4);
            int kk = d >> 2, q = d & 3, h = q & 1, s = q >> 1;
            int base = (kk * 2 + h) * 256;
            int n = c4 * 4;
            sB[base + (n + 0) * 2 + s] = v.x;
            sB[base + (n + 1) * 2 + s] = v.y;
            sB[base + (n + 2) * 2 + s] = v.z;
            sB[base + (n + 3) * 2 + s] = v.w;
        }
        __syncthreads();

        #pragma unroll
        for (int kk = 0; kk < 16; ++kk) {
            int k0 = kk * 4;
            v2f a;
            const float* arow = &sA[mrow * 832 + kbase + k0 + hi * 2];
            a.x = arow[0]; a.y = arow[1];
            v2f b0 = *(const v2f*)&sB[(kk * 2 + hi) * 256 + (nbase0 + mrow) * 2];
            v2f b1 = *(const v2f*)&sB[(kk * 2 + hi) * 256 + (nbase1 + mrow) * 2];
            acc0 = __builtin_amdgcn_wmma_f32_16x16x4_f32(false, a, false, b0,
                                                         (short)0, acc0, false, false);
            acc1 = __builtin_amdgcn_wmma_f32_16x16x4_f32(false, a, false, b1,
                                                         (short)0, acc1, false, false);
        }
        __syncthreads();
    }

    int colA = n0 + nbase0 + mrow;
    int colB = n0 + nbase1 + mrow;
    float addA = b_attn[colA], addB = b_attn[colB];
    int rowb = m0 + hi * 8;
    #pragma unroll
    for (int v = 0; v < 8; ++v) {
        rowvec[(size_t)(rowb + v) * OUT_ + colA] = acc0[v] + addA;
        rowvec[(size_t)(rowb + v) * OUT_ + colB] = acc1[v] + addB;
    }
}

// ---------------------------------------------------------------------------
// Kernel 3: ctx = sum_j p * dep ; output_sum = text + pos@Wp+bp + ctx@Wd + bd
// ---------------------------------------------------------------------------
__global__ void __launch_bounds__(256)
outsum_kernel(const float* __restrict__ text, const float* __restrict__ dep,
              const float* __restrict__ pos,  const float* __restrict__ Wd,
              const float* __restrict__ bd,   const float* __restrict__ Wp,
              const float* __restrict__ bp,   const float* __restrict__ p,
              float* __restrict__ out_sum) {
    int bi = blockIdx.x;
    int t  = threadIdx.x;  // 256
    __shared__ float sprob[L_];
    __shared__ float ctx[DEP_];
    __shared__ float spos[POS_];
    if (t < L_) sprob[t] = p[(size_t)bi * L_ + t];
    if (t >= 128 && t < 128 + POS_) spos[t - 128] = pos[(size_t)bi * POS_ + (t - 128)];
    __syncthreads();
    if (t < DEP_) {
        float acc = 0.f;
        const float* dbase = dep + (size_t)bi * L_ * DEP_ + t;
        #pragma unroll 4
        for (int j = 0; j < L_; ++j) acc += sprob[j] * dbase[(size_t)j * DEP_];
        ctx[t] = acc;
    }
    __syncthreads();
    for (int rep = 0; rep < 3; ++rep) {
        int o = t + rep * 256;
        float acc = text[(size_t)bi * D_ + o] + bd[o] + bp[o];
        #pragma unroll 4
        for (int d = 0; d < DEP_; ++d)
            acc += ctx[d] * Wd[(size_t)d * OUT_ + o] + spos[d] * Wp[(size_t)d * OUT_ + o];
        out_sum[(size_t)bi * OUT_ + o] = acc;
    }
}

// ---------------------------------------------------------------------------
// Kernel 4: r[b,i,j,o] = dep[b,i]@Wd_attn + rowvec[b,i]  via V_WMMA_F32_16X16X4_F32
// Async-DMA staging (ASYNCcnt), K-pair-swizzled B tile (ds_load_b64 fragments),
// NT-hinted streaming stores for the 402 MB r tensor (don't rinse L2).
// ---------------------------------------------------------------------------
__global__ void __launch_bounds__(128)
r_wmma_kernel(const float* __restrict__ dep, const float* __restrict__ Wd_attn,
              const float* __restrict__ rowvec, float* __restrict__ r_out) {
    int bi   = blockIdx.x;
    int t    = threadIdx.x;  // 128 = 4 waves
    int wave = t >> 5;
    int lane = t & 31;

    __shared__ float sA[L_ * DEP_];   // dep tile, 32 KB
    __shared__ float sB[DEP_ * 16];   // Wd_attn o-tile, swizzled, 4 KB
    __shared__ float sR[OUT_];        // rowvec row, 3 KB

    // async-stage dep tile and rowvec row into LDS
    const float* gA = dep + (size_t)bi * L_ * DEP_;
    for (int k = t; k < (L_ * DEP_) / 4; k += 128)       // 2048 x b128
        async_copy_b128(&sA[k * 4], gA + k * 4);
    const float* gR = rowvec + (size_t)bi * OUT_;
    for (int k = t; k < OUT_ / 4; k += 128)              // 192 x b128
        async_copy_b128(&sR[k * 4], gR + k * 4);
    wait_async();
    __syncthreads();

    float* rbase = r_out + (size_t)bi * L_ * OUT_;
    int mrow = lane & 15;
    int hi   = lane >> 4;

    for (int ot = 0; ot < 48; ++ot) {
        int o0 = ot * 16;
        // async-stage B tile swizzled: element B[d][n] at
        // sB[(kk*2+h)*32 + n*2 + s], d = 4*kk + h + 2*s
        for (int k = t; k < DEP_ * 16; k += 128) {
            int d = k >> 4, n = k & 15;
            int kk = d >> 2, q = d & 3, h = q & 1, s = q >> 1;
            int idx = (kk * 2 + h) * 32 + n * 2 + s;
            async_copy_b32(&sB[idx], Wd_attn + (size_t)d * OUT_ + o0 + n);
        }
        if (ot < 47)  // pull the next B tile toward L2/L0
            __builtin_prefetch(&Wd_attn[(size_t)(t >> 4) * OUT_ + o0 + 16 + (t & 15)], 0, 1);
        wait_async();
        __syncthreads();

        for (int rt = 0; rt < 2; ++rt) {
            int j0 = (wave + rt * 4) * 16;
            v8f acc = {};
            #pragma unroll
            for (int kk = 0; kk < 16; ++kk) {
                int k0 = kk * 4;
                const float* arow = &sA[(j0 + mrow) * DEP_ + k0 + hi * 2];
                v2f a; a.x = arow[0]; a.y = arow[1];
                v2f b = *(const v2f*)&sB[(kk * 2 + hi) * 32 + mrow * 2];
                acc = __builtin_amdgcn_wmma_f32_16x16x4_f32(
                    false, a, false, b, (short)0, acc, false, false);
            }
            int col  = o0 + mrow;
            float add = sR[col];
            int rowb = j0 + hi * 8;
            #pragma unroll
            for (int v = 0; v < 8; ++v)
                __builtin_nontemporal_store(acc[v] + add,
                    &rbase[(size_t)(rowb + v) * OUT_ + col]);
        }
        __syncthreads();
    }
}

// ---------------------------------------------------------------------------
extern "C" void kernel_launch(void* const* d_in, const int* in_sizes, int n_in,
                              void* d_out, int out_size, void* d_ws, size_t ws_size,
                              hipStream_t stream) {
    const float* text    = (const float*)d_in[0];
    const int*   adj     = (const int*)  d_in[1];
    const float* dep     = (const float*)d_in[2];
    const float* pos     = (const float*)d_in[3];
    const float* Wt      = (const float*)d_in[4];
    const float* Wp_attn = (const float*)d_in[5];
    const float* Wd_attn = (const float*)d_in[6];
    const float* b_attn  = (const float*)d_in[7];
    const float* Wd      = (const float*)d_in[8];
    const float* bd      = (const float*)d_in[9];
    const float* Wp      = (const float*)d_in[10];
    const float* bp      = (const float*)d_in[11];

    float* out    = (float*)d_out;
    float* r_out  = out;
    float* os_out = out + R_ELEMS;
    float* p_out  = os_out + OS_ELEMS;

    float* ws     = (float*)d_ws;       // 1024 floats of column sums (padded)
    float* rowvec = ws + 1024;          // B*L*OUT floats

    colsum_kernel     <<<1,       256, 0, stream>>>(Wt, Wp_attn, Wd_attn, b_attn, ws);
    softmax_kernel    <<<B_ * L_, 128, 0, stream>>>(text, adj, dep, pos, ws, p_out);
    rowvec_wmma_kernel<<<64 * 6,  128, 0, stream>>>(text, pos, Wt, Wp_attn, b_attn, rowvec);
    outsum_kernel     <<<B_ * L_, 256, 0, stream>>>(text, dep, pos, Wd, bd, Wp, bp, p_out, os_out);
    r_wmma_kernel     <<<B_ * L_, 128, 0, stream>>>(dep, Wd_attn, rowvec, r_out);
}